// TC_73349451481761
// MI455X (gfx1250) — compile-verified
//
#include <hip/hip_runtime.h>
#include <math.h>

// Problem constants (from reference): B=1024, S=128, E=256, C=256, T=12
#define BB 1024
#define SS 128
#define EE 256
#define CC 256
#define TT 12

typedef __attribute__((ext_vector_type(2))) float v2f;
typedef __attribute__((ext_vector_type(8))) float v8f;

// ---------------------------------------------------------------------------
// WMMA f32 16x16x4 fragment layout (wave32):
// A (16x4, MxK): VGPR0 = A[row][k0 + 2*kh], VGPR1 = A[row][k0 + 2*kh + 1],
//   row = lane&15, kh = lane>>4.
// B (4x16, KxN): VGPR0 = B[k0 + 2*kh][col], VGPR1 = B[k0 + 2*kh + 1][col],
//   col = lane&15. B[k][n] is K-contiguous in our layouts -> 8-byte loads.
// C/D (16x16): VGPR i = D[i + 8*kh][col].
//
// Register-blocked GEMM strip: one wave computes a 16x(8*16) output strip.
// A fragment loaded once per k-step, 8 B fragments differ only by a 16 KB
// immediate offset (j * 16 rows * 256 floats * 4 B) -> 1.125 loads per WMMA,
// and the 8 WMMAs per k-step write independent accumulators (no D->A/B RAW).
// ---------------------------------------------------------------------------
__device__ __forceinline__ void wmma_strip8(const float* __restrict__ Arow,
                                            const float* __restrict__ Bbase,
                                            int kh, v8f acc[8]) {
#pragma unroll 2
  for (int k = 0; k < CC; k += 4) {
    v2f a = *(const v2f*)(Arow + k + 2 * kh);
#pragma unroll
    for (int j = 0; j < 8; ++j) {
      v2f b = *(const v2f*)(Bbase + (size_t)j * (16 * CC) + k + 2 * kh);
      acc[j] = __builtin_amdgcn_wmma_f32_16x16x4_f32(
          /*neg_a=*/false, a, /*neg_b=*/false, b,
          /*c_mod=*/(short)0, acc[j], /*reuse_a=*/false, /*reuse_b=*/false);
    }
  }
}

// ---------------------------------------------------------------------------
// Kernel 1: pred[t][b][e] = sum_k c[b][ts][k] * Wk_w[t][e][k] + Wk_b[t][e]
// One 16x128 output strip per wave: T * 64 mblocks * 2 nhalves = 1536 waves.
// ---------------------------------------------------------------------------
__global__ void cpc_pred_kernel(const float* __restrict__ c,
                                const float* __restrict__ Wk_w,
                                const float* __restrict__ Wk_b,
                                const int* __restrict__ ts_p,
                                float* __restrict__ pred) {
  const int ts = ts_p[0];
  const int lane = threadIdx.x & 31;
  const int wave = threadIdx.x >> 5;
  const int g = blockIdx.x * 8 + wave;   // strip id in [0, 1536)
  const int t = g >> 7;                  // / (64 * 2)
  const int r = g & 127;
  const int m0 = (r >> 1) * 16;          // batch-row block
  const int nBase = (r & 1) * 128;       // E-column half (8 tiles of 16)
  const int rc = lane & 15;              // row for A, col for B
  const int kh = lane >> 4;

  const float* Arow = c + (size_t)(m0 + rc) * (SS * CC) + (size_t)ts * CC;
  const float* Bbase =
      Wk_w + (size_t)t * (EE * CC) + (size_t)(nBase + rc) * CC;

  v8f acc[8] = {};
  wmma_strip8(Arow, Bbase, kh, acc);

  const float* biasp = Wk_b + t * EE + nBase + rc;
  float* Dbase = pred + (size_t)t * (BB * EE) + nBase + rc;
#pragma unroll
  for (int j = 0; j < 8; ++j) {
    const float bias = biasp[j * 16];
#pragma unroll
    for (int i = 0; i < 8; ++i) {
      const int m = m0 + i + 8 * kh;
      Dbase[(size_t)m * EE + j * 16] = acc[j][i] + bias;
    }
  }
}

// ---------------------------------------------------------------------------
// Kernel 2: for one (t, 16-row strip): scores = enc_strip @ pred[t]^T (16x1024)
// into LDS (8 waves x one 16x128 strip each), then stable log-softmax
// reduction; emit one partial sum_r (scores[r][diag] - lse[r]) per block.
// ---------------------------------------------------------------------------
#define SROW 1040  // 1024 + 16 pad floats to spread LDS banks

__global__ void cpc_score_kernel(const float* __restrict__ z,
                                 const float* __restrict__ pred,
                                 const int* __restrict__ ts_p,
                                 float* __restrict__ partials) {
  extern __shared__ float sm[];
  float* scores = sm;                 // 16 * SROW
  float* pmax = sm + 16 * SROW;       // 256
  float* psum = pmax + 256;           // 256
  float* vals = psum + 256;           // 16

  const int ts = ts_p[0];
  const int tid = threadIdx.x;
  const int lane = tid & 31;
  const int wave = tid >> 5;
  const int t = blockIdx.x >> 6;      // / 64
  const int m0 = (blockIdx.x & 63) * 16;
  const int rc = lane & 15;
  const int kh = lane >> 4;

  // enc[t][b][e] = z[b][ts+1+t][e]
  const float* Arow =
      z + (size_t)(m0 + rc) * (SS * EE) + (size_t)(ts + 1 + t) * EE;
  // Wave's 128-column slab of pred[t] (B[k][n] = pred[t][n][k], K-contiguous)
  const int nBase = wave * 128;
  const float* Bbase =
      pred + (size_t)t * (BB * EE) + (size_t)(nBase + rc) * EE;
  __builtin_prefetch(Bbase, 0, 3);  // global_prefetch_b8 on gfx1250

  v8f acc[8] = {};
  wmma_strip8(Arow, Bbase, kh, acc);

#pragma unroll
  for (int j = 0; j < 8; ++j)
#pragma unroll
    for (int i = 0; i < 8; ++i)
      scores[(i + 8 * kh) * SROW + nBase + j * 16 + rc] = acc[j][i];
  __syncthreads();

  // Row-wise max: 16 threads per row, 64 columns each.
  {
    const int row = tid >> 4, seg = tid & 15;
    const float* rp = scores + row * SROW + seg * 64;
    float m = rp[0];
    for (int i = 1; i < 64; ++i) m = fmaxf(m, rp[i]);
    pmax[row * 16 + seg] = m;
  }
  __syncthreads();
  if (tid < 16) {
    float m = pmax[tid * 16];
    for (int i = 1; i < 16; ++i) m = fmaxf(m, pmax[tid * 16 + i]);
    pmax[tid * 16] = m;
  }
  __syncthreads();

  // Row-wise sum of exp(x - max).
  {
    const int row = tid >> 4, seg = tid & 15;
    const float m = pmax[row * 16];
    const float* rp = scores + row * SROW + seg * 64;
    float s = 0.f;
    for (int i = 0; i < 64; ++i) s += expf(rp[i] - m);
    psum[row * 16 + seg] = s;
  }
  __syncthreads();
  if (tid < 16) {
    float s = 0.f;
    for (int i = 0; i < 16; ++i) s += psum[tid * 16 + i];
    const float lse = pmax[tid * 16] + logf(s);
    // diagonal: global row b = m0 + tid, diagonal column = b
    const float diag = scores[tid * SROW + m0 + tid];
    vals[tid] = diag - lse;
  }
  __syncthreads();
  if (tid == 0) {
    float s = 0.f;
    for (int i = 0; i < 16; ++i) s += vals[i];
    partials[blockIdx.x] = s;
  }
}

// ---------------------------------------------------------------------------
// Kernel 3: out = -sum(partials) / (B * T)
// ---------------------------------------------------------------------------
__global__ void cpc_reduce_kernel(const float* __restrict__ partials, int n,
                                  float* __restrict__ out) {
  __shared__ float buf[256];
  float s = 0.f;
  for (int i = threadIdx.x; i < n; i += 256) s += partials[i];
  buf[threadIdx.x] = s;
  __syncthreads();
  for (int off = 128; off > 0; off >>= 1) {
    if (threadIdx.x < off) buf[threadIdx.x] += buf[threadIdx.x + off];
    __syncthreads();
  }
  if (threadIdx.x == 0) out[0] = -buf[0] / (float)(BB * TT);
}

// ---------------------------------------------------------------------------
extern "C" void kernel_launch(void* const* d_in, const int* in_sizes, int n_in,
                              void* d_out, int out_size, void* d_ws,
                              size_t ws_size, hipStream_t stream) {
  const float* z = (const float*)d_in[0];      // (B, S, E)
  const float* c = (const float*)d_in[1];      // (B, S, C)
  const float* Ww = (const float*)d_in[2];     // (T, E, C)
  const float* Wb = (const float*)d_in[3];     // (T, E)
  const int* ts = (const int*)d_in[4];         // scalar (device)

  float* pred = (float*)d_ws;                  // T*B*E floats = 12 MB
  float* partials = pred + (size_t)TT * BB * EE;  // 768 floats
  float* out = (float*)d_out;

  // Phase 1: pred = c_last @ Wk_w^T + Wk_b  (1536 wave strips, 8 waves/block)
  cpc_pred_kernel<<<192, 256, 0, stream>>>(c, Ww, Wb, ts, pred);

  // Phase 2: scores + log-softmax diag partials (one block per (t, 16 rows))
  const size_t smem = (16 * SROW + 256 + 256 + 16) * sizeof(float);
  cpc_score_kernel<<<TT * 64, 256, smem, stream>>>(z, pred, ts, partials);

  // Phase 3: final scalar
  cpc_reduce_kernel<<<1, 256, 0, stream>>>(partials, TT * 64, out);
}